// AdaptiveTripletLoss_28681791603354
// MI455X (gfx1250) — compile-verified
//
#include <hip/hip_runtime.h>
#include <hip/hip_bf16.h>
#include <math.h>

typedef __attribute__((ext_vector_type(2))) float v2f;
typedef __attribute__((ext_vector_type(8))) float v8f;
typedef __attribute__((ext_vector_type(4))) int   v4i;

typedef v4i __attribute__((address_space(1))) g_v4i;   // global
typedef v4i __attribute__((address_space(3))) l_v4i;   // LDS

#define NS 4096
#define DIM 512
#define NCLS 64
#define APAD 516   // 16-row A tile stride (mod 64 banks = 4 -> conflict-light)

#if __has_builtin(__builtin_amdgcn_global_load_async_to_lds_b128)
#define HAVE_ASYNC_LDS 1
#endif

__device__ __forceinline__ float fast_sqrtf(float x) {
#if __has_builtin(__builtin_amdgcn_sqrtf)
    return __builtin_amdgcn_sqrtf(x);   // raw v_sqrt_f32, ~1 ulp, no NR refinement
#else
    return sqrtf(x);
#endif
}

// ---------------- k0: zero the atomically-accumulated workspace ----------------
__global__ void k_zero(float* p, int n) {
    int i = blockIdx.x * blockDim.x + threadIdx.x;
    if (i < n) p[i] = 0.0f;
}

// ---------------- k1: per-sample squared norm + per-class sums ----------------
__global__ void __launch_bounds__(256)
k_stats(const float* __restrict__ feats, const int* __restrict__ labels,
        float* __restrict__ sq, float* __restrict__ counts,
        float* __restrict__ csum, float* __restrict__ csumsq) {
    const int n = blockIdx.x, tid = threadIdx.x;
    const int lab = labels[n];
    const float* f = feats + (size_t)n * DIM;
    float s = 0.0f;
    for (int d = tid; d < DIM; d += 256) {
        float v = f[d];
        s += v * v;
        atomicAdd(&csum[lab * DIM + d], v);
        atomicAdd(&csumsq[lab * DIM + d], v * v);
    }
    __shared__ float red[256];
    red[tid] = s; __syncthreads();
    for (int st = 128; st > 0; st >>= 1) { if (tid < st) red[tid] += red[tid + st]; __syncthreads(); }
    if (tid == 0) { sq[n] = red[0]; atomicAdd(&counts[lab], 1.0f); }
}

// ---------------- k2: class mean (in place into csum) + mean-of-variance ----------------
__global__ void __launch_bounds__(256)
k_classfinal(const float* __restrict__ counts, float* __restrict__ csum,
             const float* __restrict__ csumsq, float* __restrict__ meanvar) {
    const int c = blockIdx.x, tid = threadIdx.x;
    const float denom = fmaxf(counts[c], 1.0f);
    float vs = 0.0f;
    for (int d = tid; d < DIM; d += 256) {
        float m  = csum[c * DIM + d] / denom;
        float mq = csumsq[c * DIM + d] / denom;
        vs += fmaxf(mq - m * m, 0.0f);
        csum[c * DIM + d] = m;   // cmean
    }
    __shared__ float red[256];
    red[tid] = vs; __syncthreads();
    for (int st = 128; st > 0; st >>= 1) { if (tid < st) red[tid] += red[tid + st]; __syncthreads(); }
    if (tid == 0) meanvar[c] = red[0] / (float)DIM;
}

// ---------------- k3: stat margin = ||f - mean[y]|| * meanvar[y] ----------------
__global__ void __launch_bounds__(256)
k_statmargin(const float* __restrict__ feats, const int* __restrict__ labels,
             const float* __restrict__ cmean, const float* __restrict__ meanvar,
             float* __restrict__ statm) {
    const int n = blockIdx.x, tid = threadIdx.x;
    const int lab = labels[n];
    float s = 0.0f;
    for (int d = tid; d < DIM; d += 256) {
        float df = feats[(size_t)n * DIM + d] - cmean[lab * DIM + d];
        s = fmaf(df, df, s);
    }
    __shared__ float red[256];
    red[tid] = s; __syncthreads();
    for (int st = 128; st > 0; st >>= 1) { if (tid < st) red[tid] += red[tid + st]; __syncthreads(); }
    if (tid == 0) {
        float cd = red[0] > 0.0f ? fast_sqrtf(red[0]) : 0.0f;
        statm[n] = cd * meanvar[lab];
    }
}

// ---------------- k4: fused WMMA Gram-tile -> distance -> per-row reductions ----------------
__global__ void __launch_bounds__(256)
k_pairwise(const float* __restrict__ feats, const int* __restrict__ labels,
           const float* __restrict__ sq,
           float* __restrict__ hpos, float* __restrict__ hneg,
           float* __restrict__ spos, float* __restrict__ sneg,
           float* __restrict__ pcnt, float* __restrict__ ncnt) {
    const int rowbase = blockIdx.x * 16;
    const int tid  = threadIdx.x;
    const int wave = tid >> 5;
    const int lane = tid & 31;
    const int mh   = lane & 15;          // row-within-tile for A/B fragments, N for C/D
    const int half = lane >> 4;          // 0 or 1
    const int ko   = half * 2;           // K sub-offset per VGPR layout (16x4 f32 A/B)
    const int rh   = half * 8;           // C/D: vgpr r -> matrix row r + rh

    __shared__ __align__(16) float As[16 * APAD];
    __shared__ float sqrow_s[16];
    __shared__ int   labrow_s[16];
    __shared__ float red[8][16][6];

    // Stage 16x512 A tile (32 KB) into LDS.
#ifdef HAVE_ASYNC_LDS
    for (int i = tid; i < 16 * (DIM / 4); i += 256) {
        int m  = i >> 7;            // DIM/4 = 128 float4 per row
        int kk = (i & 127) << 2;
        __builtin_amdgcn_global_load_async_to_lds_b128(
            (g_v4i*)(feats + (size_t)(rowbase + m) * DIM + kk),
            (l_v4i*)(&As[m * APAD + kk]),
            0, 0);
    }
    asm volatile("s_wait_asynccnt 0x0" ::: "memory");
#else
    for (int i = tid; i < 16 * (DIM / 4); i += 256) {
        int m  = i >> 7;
        int kk = (i & 127) << 2;
        float4 v = *(const float4*)(feats + (size_t)(rowbase + m) * DIM + kk);
        *(float4*)(&As[m * APAD + kk]) = v;
    }
#endif
    if (tid < 16) { sqrow_s[tid] = sq[rowbase + tid]; labrow_s[tid] = labels[rowbase + tid]; }
    __syncthreads();

    float sq_r[8]; int lab_r[8];
#pragma unroll
    for (int r = 0; r < 8; ++r) { sq_r[r] = sqrow_s[r + rh]; lab_r[r] = labrow_s[r + rh]; }

    float a_hp[8], a_hn[8], a_sp[8], a_sn[8], a_pc[8], a_nc[8];
#pragma unroll
    for (int r = 0; r < 8; ++r) { a_hp[r] = -1e30f; a_hn[r] = 1e30f; a_sp[r] = 0.f; a_sn[r] = 0.f; a_pc[r] = 0.f; a_nc[r] = 0.f; }

    const float* Arow = &As[mh * APAD];

    // branchless fold of one 16x16 distance tile held in a v8f fragment
    auto fold_tile = [&](const float* cf, float sq_c, int lab_c, int gcol) {
#pragma unroll
        for (int r = 0; r < 8; ++r) {
            float d2   = fmaxf(fmaf(-2.0f, cf[r], sq_r[r] + sq_c), 0.0f);
            float dist = fast_sqrtf(d2);
            const int  grow  = rowbase + rh + r;
            const bool same  = (lab_r[r] == lab_c);
            const bool isPos = same && (grow != gcol);
            const bool isNeg = !same;
            a_hp[r] = fmaxf(a_hp[r], isPos ? dist : -1e30f);
            a_hn[r] = fminf(a_hn[r], isNeg ? dist :  1e30f);
            a_sp[r] += isPos ? dist : 0.0f;
            a_sn[r] += isNeg ? dist : 0.0f;
            a_pc[r] += isPos ? 1.0f : 0.0f;
            a_nc[r] += isNeg ? 1.0f : 0.0f;
        }
    };

    // Each wave handles two adjacent 16-col tiles per iteration: A fragment reused 2x.
    for (int jp = wave; jp < NS / 32; jp += 8) {
        const int colbase = jp * 32;
        const float* B0 = feats + (size_t)(colbase      + mh) * DIM;
        const float* B1 = feats + (size_t)(colbase + 16 + mh) * DIM;
        const float sq_c0 = sq[colbase + mh],      sq_c1 = sq[colbase + 16 + mh];
        const int  lab_c0 = labels[colbase + mh],  lab_c1 = labels[colbase + 16 + mh];
        if (jp + 8 < NS / 32) {  // wave-uniform prefetch of next pair's B rows
            __builtin_prefetch(feats + (size_t)(colbase + 256 + mh) * DIM, 0, 1);
            __builtin_prefetch(feats + (size_t)(colbase + 272 + mh) * DIM, 0, 1);
        }

        union { v8f v; float f[8]; } c0, c1;
        c0.v = (v8f){};
        c1.v = (v8f){};
#pragma unroll 4
        for (int kc = 0; kc < DIM; kc += 4) {
            v2f a, b0, b1;
            a.x  = Arow[kc + ko]; a.y  = Arow[kc + ko + 1];
            b0.x = B0[kc + ko];   b0.y = B0[kc + ko + 1];
            b1.x = B1[kc + ko];   b1.y = B1[kc + ko + 1];
            c0.v = __builtin_amdgcn_wmma_f32_16x16x4_f32(
                false, a, false, b0, (short)0, c0.v, false, false);
            c1.v = __builtin_amdgcn_wmma_f32_16x16x4_f32(
                false, a, false, b1, (short)0, c1.v, false, false);
        }
        fold_tile(c0.f, sq_c0, lab_c0, colbase + mh);
        fold_tile(c1.f, sq_c1, lab_c1, colbase + 16 + mh);
    }

    // reduce across the 16 lanes of each half (xor 1,2,4,8 stays within the half)
#pragma unroll
    for (int off = 1; off < 16; off <<= 1) {
#pragma unroll
        for (int r = 0; r < 8; ++r) {
            a_hp[r] = fmaxf(a_hp[r], __shfl_xor(a_hp[r], off, 32));
            a_hn[r] = fminf(a_hn[r], __shfl_xor(a_hn[r], off, 32));
            a_sp[r] += __shfl_xor(a_sp[r], off, 32);
            a_sn[r] += __shfl_xor(a_sn[r], off, 32);
            a_pc[r] += __shfl_xor(a_pc[r], off, 32);
            a_nc[r] += __shfl_xor(a_nc[r], off, 32);
        }
    }
    if ((lane & 15) == 0) {
#pragma unroll
        for (int r = 0; r < 8; ++r) {
            red[wave][rh + r][0] = a_hp[r];
            red[wave][rh + r][1] = a_hn[r];
            red[wave][rh + r][2] = a_sp[r];
            red[wave][rh + r][3] = a_sn[r];
            red[wave][rh + r][4] = a_pc[r];
            red[wave][rh + r][5] = a_nc[r];
        }
    }
    __syncthreads();
    if (tid < 16) {
        float hp = -1e30f, hn = 1e30f, sp = 0.f, sn = 0.f, pc = 0.f, nc = 0.f;
        for (int w = 0; w < 8; ++w) {
            hp = fmaxf(hp, red[w][tid][0]);
            hn = fminf(hn, red[w][tid][1]);
            sp += red[w][tid][2]; sn += red[w][tid][3];
            pc += red[w][tid][4]; nc += red[w][tid][5];
        }
        const int row = rowbase + tid;
        hpos[row] = hp; hneg[row] = hn; spos[row] = sp;
        sneg[row] = sn; pcnt[row] = pc; ncnt[row] = nc;
    }
}

// ---------------- k5: final scalar loss ----------------
__global__ void __launch_bounds__(256)
k_final(const float* __restrict__ hpos, const float* __restrict__ hneg,
        const float* __restrict__ spos, const float* __restrict__ sneg,
        const float* __restrict__ pcnt, const float* __restrict__ ncnt,
        const float* __restrict__ statm, float* __restrict__ out) {
    const int tid = threadIdx.x;
    float lsum = 0.0f, lcnt = 0.0f;
    for (int n = tid; n < NS; n += 256) {
        float pc = pcnt[n], nc = ncnt[n];
        float mp = spos[n] / fmaxf(pc, 1.0f);
        float mn = sneg[n] / fmaxf(nc, 1.0f);
        float margin = 0.1f + 0.1f * (mn - mp) + 0.1f * statm[n];
        float per = fmaxf(hpos[n] - hneg[n] + margin, 0.0f);
        float valid = (pc > 0.0f && nc > 0.0f) ? 1.0f : 0.0f;
        lsum += valid * per;
        lcnt += valid;
    }
    __shared__ float rs[256], rc[256];
    rs[tid] = lsum; rc[tid] = lcnt; __syncthreads();
    for (int st = 128; st > 0; st >>= 1) {
        if (tid < st) { rs[tid] += rs[tid + st]; rc[tid] += rc[tid + st]; }
        __syncthreads();
    }
    if (tid == 0) out[0] = (rc[0] > 0.0f) ? (rs[0] / fmaxf(rc[0], 1.0f)) : 0.0f;
}

// ---------------- host ----------------
extern "C" void kernel_launch(void* const* d_in, const int* in_sizes, int n_in,
                              void* d_out, int out_size, void* d_ws, size_t ws_size,
                              hipStream_t stream) {
    const float* feats = (const float*)d_in[0];
    const int* labels  = (const int*)d_in[1];
    float* out = (float*)d_out;

    float* ws      = (float*)d_ws;
    float* sq      = ws;                       // NS
    float* counts  = sq + NS;                  // NCLS
    float* csum    = counts + NCLS;            // NCLS*DIM  (becomes cmean in k2)
    float* csumsq  = csum + NCLS * DIM;        // NCLS*DIM
    float* meanvar = csumsq + NCLS * DIM;      // NCLS
    float* statm   = meanvar + NCLS;           // NS
    float* hpos    = statm + NS;               // NS
    float* hneg    = hpos + NS;                // NS
    float* spos    = hneg + NS;                // NS
    float* sneg    = spos + NS;                // NS
    float* pcnt    = sneg + NS;                // NS
    float* ncnt    = pcnt + NS;                // NS

    const int nzero = NCLS + 2 * NCLS * DIM;   // counts + csum + csumsq
    k_zero<<<(nzero + 255) / 256, 256, 0, stream>>>(counts, nzero);
    k_stats<<<NS, 256, 0, stream>>>(feats, labels, sq, counts, csum, csumsq);
    k_classfinal<<<NCLS, 256, 0, stream>>>(counts, csum, csumsq, meanvar);
    k_statmargin<<<NS, 256, 0, stream>>>(feats, labels, csum, meanvar, statm);
    k_pairwise<<<NS / 16, 256, 0, stream>>>(feats, labels, sq,
                                            hpos, hneg, spos, sneg, pcnt, ncnt);
    k_final<<<1, 256, 0, stream>>>(hpos, hneg, spos, sneg, pcnt, ncnt, statm, out);
}